// LightGCNEncoder_49486613185211
// MI455X (gfx1250) — compile-verified
//
#include <hip/hip_runtime.h>
#include <stdint.h>

// ---------------- problem constants (match reference) ----------------
#define N_NODES 100000
#define D       32
#define TILE    2048          // edges per TDM tile (3 * 8KB per stage buffer)
#define SPMM_BLOCKS  256
#define SPMM_THREADS 256

// ---------------- TDM descriptor types (6-arg builtin, clang-23 lane) ----
typedef __attribute__((ext_vector_type(4))) unsigned int v4u;
typedef __attribute__((ext_vector_type(8))) int          v8i;
typedef __attribute__((ext_vector_type(4))) int          v4i;

// Issue a 1-D TDM load of n 4-byte elements: global -> LDS.
// D# per CDNA5 ISA ch.8: group0 = {count=1 | lds_addr | global_addr | type=2},
// group1 = {wg_mask=0, data_size=2(4B), tensor_dim0=n, tensor_dim1=1,
//           tile_dim0=n, tile_dim1/2=0, dim0_stride=n}. Groups 2/3 = 0 (<=2D).
__device__ __forceinline__ void tdm_load_1d(const void* gptr, unsigned lds_off,
                                            unsigned n) {
  unsigned long long ga = (unsigned long long)(uintptr_t)gptr;
  v4u g0;
  g0[0] = 1u;                                           // count=1, is_restore=0
  g0[1] = lds_off;                                      // LDS byte address
  g0[2] = (unsigned)(ga & 0xFFFFFFFFu);                 // global_addr[31:0]
  g0[3] = (unsigned)((ga >> 32) & 0x01FFFFFFu)          // global_addr[56:32]
          | 0x80000000u;                                // type=2 ("image")
  v8i g1;
  g1[0] = (int)(2u << 16);                              // data_size=4B, mask=0
  g1[1] = (int)((n & 0xFFFFu) << 16);                   // tensor_dim0[15:0]
  g1[2] = (int)(((n >> 16) & 0xFFFFu) | (1u << 16));    // dim0[31:16], dim1=1
  g1[3] = (int)((n & 0xFFFFu) << 16);                   // tile_dim0 = n
  g1[4] = 0;                                            // tile_dim1/2 unused
  g1[5] = (int)n;                                       // dim0_stride lo
  g1[6] = 0;
  g1[7] = 0;
  v4i z4 = {0, 0, 0, 0};
  v8i z8 = {0, 0, 0, 0, 0, 0, 0, 0};
  __builtin_amdgcn_tensor_load_to_lds(g0, g1, z4, z4, z8, 0);
}

// ---------------- SpMM: h_next += A * h_cur (h_next pre-zeroed) --------
// 8 lanes per edge; lane-chunk c covers features [4c, 4c+4). Edge streams are
// staged into LDS by the TDM with double buffering (wave 0 drives the DMA).
__global__ __launch_bounds__(SPMM_THREADS) void spmm_tdm_kernel(
    const int* __restrict__ rows, const int* __restrict__ cols,
    const float* __restrict__ vals, int nedges,
    const float* __restrict__ hcur, float* __restrict__ hnext) {
  __shared__ int   s_row[2][TILE];
  __shared__ int   s_col[2][TILE];
  __shared__ float s_val[2][TILE];

  const int  tid    = threadIdx.x;
  const bool leader = (tid < 32);                 // wave 0 owns the TDM
  const int  ntiles = (nedges + TILE - 1) / TILE;

  int tile   = (int)blockIdx.x;
  int parity = 0;

  if (tile < ntiles && leader) {
    const int base = tile * TILE;
    const unsigned n = (unsigned)min(TILE, nedges - base);
    tdm_load_1d(rows + base, (unsigned)(uintptr_t)&s_row[0][0], n);
    tdm_load_1d(cols + base, (unsigned)(uintptr_t)&s_col[0][0], n);
    tdm_load_1d(vals + base, (unsigned)(uintptr_t)&s_val[0][0], n);
  }

  const float4* __restrict__ hc4 = reinterpret_cast<const float4*>(hcur);

  for (; tile < ntiles; tile += (int)gridDim.x) {
    if (leader) __builtin_amdgcn_s_wait_tensorcnt(0);  // buf[parity] landed
    __syncthreads();

    const int next_tile = tile + (int)gridDim.x;       // prefetch next tile
    if (next_tile < ntiles && leader) {
      const int base = next_tile * TILE;
      const unsigned n = (unsigned)min(TILE, nedges - base);
      const int p = parity ^ 1;
      tdm_load_1d(rows + base, (unsigned)(uintptr_t)&s_row[p][0], n);
      tdm_load_1d(cols + base, (unsigned)(uintptr_t)&s_col[p][0], n);
      tdm_load_1d(vals + base, (unsigned)(uintptr_t)&s_val[p][0], n);
    }

    const int n    = min(TILE, nedges - tile * TILE);
    const int work = n * 8;                            // (edge, 4-feat chunk)
    for (int k = tid; k < work; k += SPMM_THREADS) {
      const int   e = k >> 3;
      const int   c = k & 7;
      const int   r = s_row[parity][e];
      const int   g = s_col[parity][e];
      const float v = s_val[parity][e];
      const float4 h = hc4[g * (D / 4) + c];           // global_load_b128
      float* dst = hnext + (r * D + c * 4);
      unsafeAtomicAdd(dst + 0, v * h.x);               // global_atomic_add_f32
      unsafeAtomicAdd(dst + 1, v * h.y);
      unsafeAtomicAdd(dst + 2, v * h.z);
      unsafeAtomicAdd(dst + 3, v * h.w);
    }
    __syncthreads();                                   // done reading buf
    parity ^= 1;
  }
}

// ---------------- aux kernels (float4, memory-bound trivia) -------------
__global__ void lgcn_init_kernel(const float4* __restrict__ x,
                                 float4* __restrict__ b0,
                                 float4* __restrict__ b1,
                                 float4* __restrict__ acc, int n4) {
  const int i = blockIdx.x * blockDim.x + threadIdx.x;
  if (i < n4) {
    const float4 v = x[i];
    b0[i]  = v;                                  // layer-0 embeddings
    acc[i] = v;                                  // running sum starts at x
    b1[i]  = make_float4(0.f, 0.f, 0.f, 0.f);    // first SpMM target
  }
}

// acc = (acc + nxt) * scale; optionally zero the buffer that becomes the
// next SpMM destination (fuses the per-layer memset).
__global__ void lgcn_accum_kernel(float4* __restrict__ acc,
                                  const float4* __restrict__ nxt,
                                  float4* __restrict__ to_zero,
                                  float scale, int n4) {
  const int i = blockIdx.x * blockDim.x + threadIdx.x;
  if (i < n4) {
    float4 a = acc[i];
    const float4 b = nxt[i];
    a.x = (a.x + b.x) * scale;
    a.y = (a.y + b.y) * scale;
    a.z = (a.z + b.z) * scale;
    a.w = (a.w + b.w) * scale;
    acc[i] = a;
    if (to_zero) to_zero[i] = make_float4(0.f, 0.f, 0.f, 0.f);
  }
}

// ---------------- driver ------------------------------------------------
extern "C" void kernel_launch(void* const* d_in, const int* in_sizes, int n_in,
                              void* d_out, int out_size, void* d_ws,
                              size_t ws_size, hipStream_t stream) {
  const int*   rows = (const int*)d_in[0];
  const int*   cols = (const int*)d_in[1];
  const float* vals = (const float*)d_in[2];
  const float* x    = (const float*)d_in[3];
  const int    E    = in_sizes[0];

  float* acc = (float*)d_out;                       // [N, D] accumulator
  float* b0  = (float*)d_ws;                        // ping
  float* b1  = b0 + (size_t)N_NODES * D;            // pong

  const int n4 = N_NODES * D / 4;
  const dim3 ablk(256), agrd((n4 + 255) / 256);

  lgcn_init_kernel<<<agrd, ablk, 0, stream>>>(
      (const float4*)x, (float4*)b0, (float4*)b1, (float4*)acc, n4);

  // layer 1: b1 = A*b0 ; acc += b1 ; zero b0 (next target)
  spmm_tdm_kernel<<<SPMM_BLOCKS, SPMM_THREADS, 0, stream>>>(rows, cols, vals,
                                                            E, b0, b1);
  lgcn_accum_kernel<<<agrd, ablk, 0, stream>>>((float4*)acc, (const float4*)b1,
                                               (float4*)b0, 1.0f, n4);
  // layer 2: b0 = A*b1 ; acc += b0 ; zero b1 (next target)
  spmm_tdm_kernel<<<SPMM_BLOCKS, SPMM_THREADS, 0, stream>>>(rows, cols, vals,
                                                            E, b1, b0);
  lgcn_accum_kernel<<<agrd, ablk, 0, stream>>>((float4*)acc, (const float4*)b0,
                                               (float4*)b1, 1.0f, n4);
  // layer 3: b1 = A*b0 ; acc = (acc + b1) / 4
  spmm_tdm_kernel<<<SPMM_BLOCKS, SPMM_THREADS, 0, stream>>>(rows, cols, vals,
                                                            E, b0, b1);
  lgcn_accum_kernel<<<agrd, ablk, 0, stream>>>((float4*)acc, (const float4*)b1,
                                               (float4*)nullptr, 0.25f, n4);
}